// GraphRotationTransform_7567732375972
// MI455X (gfx1250) — compile-verified
//
#include <hip/hip_runtime.h>
#include <math.h>

// Clang ext-vector types: required for __builtin_nontemporal_load/store and
// guaranteed single b128 VMEM ops.
typedef float f4 __attribute__((ext_vector_type(4)));
typedef int   i4 __attribute__((ext_vector_type(4)));

#define TWO_PI_F 6.28318530717958647692f
#define BS       256   // 8 wave32 per block
#define RPT      4     // rows per thread (48 B -> 3x b128)
#define LDS_CAP  256   // staged groups per block: 256 * 48 B = 12 KB LDS

// Explicit address-space pointer types: prevent the compiler from tail-merging
// the LDS and global R-read branches into generic (flat) loads. AS3 deref must
// lower to ds_load_b128; AS1 deref must lower to global_load_b128.
typedef __attribute__((address_space(1))) i4*       gbl_i4p;
typedef __attribute__((address_space(3))) i4*       lds_i4p;
typedef const __attribute__((address_space(1))) f4* gbl_cf4p;
typedef const __attribute__((address_space(3))) f4* lds_cf4p;

// ---------------------------------------------------------------------------
// gfx1250 async global->LDS staging (ASYNCcnt path), with safe fallbacks.
// Builtin signature (verified by hipcc): (v4i AS1* src, v4i AS3* dst, Imm, Imm).
// ---------------------------------------------------------------------------
#if __has_builtin(__builtin_amdgcn_global_load_async_to_lds_b128)
#define HAVE_ASYNC_LDS 1
#else
#define HAVE_ASYNC_LDS 0
#endif

__device__ __forceinline__ void stage16(const float* __restrict__ gsrc,
                                        float* lds_dst) {
#if HAVE_ASYNC_LDS
    __builtin_amdgcn_global_load_async_to_lds_b128(
        (gbl_i4p)gsrc, (lds_i4p)lds_dst, /*offset=*/0, /*cpol=*/0);
#else
    *(lds_i4p)lds_dst = *(gbl_i4p)gsrc;   // synchronous fallback (still LDS)
#endif
}

__device__ __forceinline__ void stage_wait() {
#if HAVE_ASYNC_LDS
#if __has_builtin(__builtin_amdgcn_s_wait_asynccnt)
    __builtin_amdgcn_s_wait_asynccnt(0);
#else
    asm volatile("s_wait_asynccnt 0x0" ::: "memory");
#endif
#endif
}

// ---------------------------------------------------------------------------
// Rodrigues rotation rows (R = cos I + sin K + (1-cos) a a^T), identity when
// apply_rand >= 0.5. Row j gives out_j = sum_i R[j][i] v_i  ('ni,nji->nj').
// ---------------------------------------------------------------------------
__device__ __forceinline__ f4 mk4(float a, float b, float c, float d) {
    f4 v; v.x = a; v.y = b; v.z = c; v.w = d; return v;
}

__device__ __forceinline__ void rodrigues_rows(float ax, float ay, float az,
                                               float ang, float apply,
                                               f4& R0, f4& R1, f4& R2) {
    if (apply < 0.5f) {
        float inv = 1.0f / sqrtf(ax * ax + ay * ay + az * az);
        float a0 = ax * inv, a1 = ay * inv, a2 = az * inv;
        float s, c;
        __sincosf(ang * TWO_PI_F, &s, &c);
        float t = 1.0f - c;
        R0 = mk4(fmaf(t * a0, a0, c), fmaf(t * a0, a1, -s * a2), fmaf(t * a0, a2,  s * a1), 0.0f);
        R1 = mk4(fmaf(t * a0, a1,  s * a2), fmaf(t * a1, a1, c), fmaf(t * a1, a2, -s * a0), 0.0f);
        R2 = mk4(fmaf(t * a0, a2, -s * a1), fmaf(t * a1, a2,  s * a0), fmaf(t * a2, a2, c), 0.0f);
    } else {
        R0 = mk4(1.0f, 0.0f, 0.0f, 0.0f);
        R1 = mk4(0.0f, 1.0f, 0.0f, 0.0f);
        R2 = mk4(0.0f, 0.0f, 1.0f, 0.0f);
    }
}

// ---------------------------------------------------------------------------
// Pass 1: build R table, padded to 12 floats (48 B) per group so every gather
// is three aligned b128 ops. 8192 * 48 B = 384 KB -> permanently L2-resident.
// ---------------------------------------------------------------------------
__global__ void build_R_kernel(const float* __restrict__ axis,
                               const float* __restrict__ angle,
                               const float* __restrict__ apply_rand,
                               float* __restrict__ Rtab, int G) {
    int g = blockIdx.x * blockDim.x + threadIdx.x;
    if (g >= G) return;
    f4 R0, R1, R2;
    rodrigues_rows(axis[3 * g], axis[3 * g + 1], axis[3 * g + 2],
                   angle[g], apply_rand[g], R0, R1, R2);
    f4* op = (f4*)(Rtab + (size_t)g * 12);
    op[0] = R0; op[1] = R1; op[2] = R2;
}

// ---------------------------------------------------------------------------
// Pass 2: streaming rotate. 4 rows/thread => 3x b128 TH_NT load, 3x b128
// TH_NT store, 1x b128 index load. USE_LDS: block stages its (sorted, hence
// contiguous & tiny) group window of the R table into LDS with the async DMA
// path, so per-row R reads are ds_load_b128 and the VMEM port is dedicated to
// the 432 MB stream. RECOMPUTE: rebuild R per row if d_ws is too small.
// ---------------------------------------------------------------------------
template <bool USE_LDS, bool RECOMPUTE>
__global__ void rot_apply_kernel(const float* __restrict__ in,
                                 const int* __restrict__ idx,   // null => g = row
                                 const float* __restrict__ Rtab,
                                 const float* __restrict__ axis,
                                 const float* __restrict__ angle,
                                 const float* __restrict__ apply_rand,
                                 float* __restrict__ out, int n) {
    extern __shared__ float sR[];  // LDS_CAP * 12 floats when USE_LDS
    const int tid       = threadIdx.x;
    const int blockBase = blockIdx.x * (BS * RPT);

    int gLo = 0, span = 0;
    if (USE_LDS) {
        int lastRow = blockBase + BS * RPT;
        if (lastRow > n) lastRow = n;
        lastRow -= 1;
        gLo = idx[blockBase];            // same-address load -> L2 broadcast
        const int gHi = idx[lastRow];
        span = gHi - gLo + 1;
        if (span > LDS_CAP) span = LDS_CAP;
        for (int j = tid; j < span * 3; j += BS)
            stage16(Rtab + (size_t)gLo * 12 + (size_t)j * 4, &sR[j * 4]);
        stage_wait();                    // drain this wave's ASYNCcnt
        __syncthreads();                 // LDS visible to all waves
    }

    const int r0 = blockBase + tid * RPT;
    if (r0 >= n) return;

    auto loadR = [&](int g, f4& A, f4& B, f4& C) {
        if (RECOMPUTE) {
            rodrigues_rows(axis[3 * g], axis[3 * g + 1], axis[3 * g + 2],
                           angle[g], apply_rand[g], A, B, C);
        } else if (USE_LDS && (unsigned)(g - gLo) < (unsigned)span) {
            lds_cf4p rp = (lds_cf4p)(sR + (size_t)(g - gLo) * 12);
            A = rp[0]; B = rp[1]; C = rp[2];          // ds_load_b128 x3
        } else {
            gbl_cf4p rp = (gbl_cf4p)(Rtab + (size_t)g * 12);
            A = rp[0]; B = rp[1]; C = rp[2];          // L2-resident fallback
        }
    };

    if (r0 + RPT <= n) {
        const f4* ip = (const f4*)(in + (size_t)r0 * 3);
        f4 d0 = __builtin_nontemporal_load(ip + 0);
        f4 d1 = __builtin_nontemporal_load(ip + 1);
        f4 d2 = __builtin_nontemporal_load(ip + 2);

        int gg[4];
        if (idx) {
            i4 gi = __builtin_nontemporal_load((const i4*)(idx + r0));
            gg[0] = gi.x; gg[1] = gi.y; gg[2] = gi.z; gg[3] = gi.w;
        } else {
            gg[0] = r0; gg[1] = r0 + 1; gg[2] = r0 + 2; gg[3] = r0 + 3;
        }

        float vx[4] = { d0.x, d0.w, d1.z, d2.y };
        float vy[4] = { d0.y, d1.x, d1.w, d2.z };
        float vz[4] = { d0.z, d1.y, d2.x, d2.w };
        float ox[4], oy[4], oz[4];

        f4 A = mk4(0, 0, 0, 0), B = A, C = A;
        int gprev = -1;
#pragma unroll
        for (int k = 0; k < 4; ++k) {
            if (gg[k] != gprev) {        // sorted batches: usually 1 R / thread
                loadR(gg[k], A, B, C);
                gprev = gg[k];
            }
            ox[k] = fmaf(A.x, vx[k], fmaf(A.y, vy[k], A.z * vz[k]));
            oy[k] = fmaf(B.x, vx[k], fmaf(B.y, vy[k], B.z * vz[k]));
            oz[k] = fmaf(C.x, vx[k], fmaf(C.y, vy[k], C.z * vz[k]));
        }

        f4 e0 = mk4(ox[0], oy[0], oz[0], ox[1]);
        f4 e1 = mk4(oy[1], oz[1], ox[2], oy[2]);
        f4 e2 = mk4(oz[2], ox[3], oy[3], oz[3]);
        f4* op = (f4*)(out + (size_t)r0 * 3);
        __builtin_nontemporal_store(e0, op + 0);
        __builtin_nontemporal_store(e1, op + 1);
        __builtin_nontemporal_store(e2, op + 2);
    } else {
        for (int r = r0; r < n; ++r) {   // scalar tail (normally dead)
            int g = idx ? idx[r] : r;
            f4 A, B, C;
            loadR(g, A, B, C);
            float x = in[3 * r], y = in[3 * r + 1], z = in[3 * r + 2];
            out[3 * r]     = fmaf(A.x, x, fmaf(A.y, y, A.z * z));
            out[3 * r + 1] = fmaf(B.x, x, fmaf(B.y, y, B.z * z));
            out[3 * r + 2] = fmaf(C.x, x, fmaf(C.y, y, C.z * z));
        }
    }
}

// ---------------------------------------------------------------------------
extern "C" void kernel_launch(void* const* d_in, const int* in_sizes, int n_in,
                              void* d_out, int out_size, void* d_ws, size_t ws_size,
                              hipStream_t stream) {
    (void)n_in; (void)out_size;

    const float* x     = (const float*)d_in[0];
    const float* ea    = (const float*)d_in[1];
    const float* y     = (const float*)d_in[2];
    const int*   nb    = (const int*)d_in[3];
    const int*   eb    = (const int*)d_in[4];
    const float* axis  = (const float*)d_in[5];
    const float* angle = (const float*)d_in[6];
    const float* apr   = (const float*)d_in[7];

    const int N = in_sizes[0] / 3;
    const int E = in_sizes[1] / 3;
    const int G = in_sizes[2] / 3;

    float* out = (float*)d_out;
    float* xo  = out;
    float* eo  = out + (size_t)N * 3;
    float* yo  = out + ((size_t)N + (size_t)E) * 3;

    auto gridFor = [](int n) { int nv = (n + RPT - 1) / RPT; return (nv + BS - 1) / BS; };
    const size_t ldsBytes   = (size_t)LDS_CAP * 12 * sizeof(float);
    const size_t rtab_bytes = (size_t)G * 12 * sizeof(float);

    if (ws_size >= rtab_bytes) {
        float* Rtab = (float*)d_ws;
        build_R_kernel<<<(G + BS - 1) / BS, BS, 0, stream>>>(axis, angle, apr, Rtab, G);
        rot_apply_kernel<true, false><<<gridFor(N), BS, ldsBytes, stream>>>(
            x, nb, Rtab, nullptr, nullptr, nullptr, xo, N);
        rot_apply_kernel<true, false><<<gridFor(E), BS, ldsBytes, stream>>>(
            ea, eb, Rtab, nullptr, nullptr, nullptr, eo, E);
        rot_apply_kernel<false, false><<<gridFor(G), BS, 0, stream>>>(
            y, nullptr, Rtab, nullptr, nullptr, nullptr, yo, G);
    } else {
        // Fallback: recompute R per row (axis/angle tables tiny, L2-resident).
        rot_apply_kernel<false, true><<<gridFor(N), BS, 0, stream>>>(
            x, nb, nullptr, axis, angle, apr, xo, N);
        rot_apply_kernel<false, true><<<gridFor(E), BS, 0, stream>>>(
            ea, eb, nullptr, axis, angle, apr, eo, E);
        rot_apply_kernel<false, true><<<gridFor(G), BS, 0, stream>>>(
            y, nullptr, nullptr, axis, angle, apr, yo, G);
    }
}